// DenseGINELayer_78116865179706
// MI455X (gfx1250) — compile-verified
//
#include <hip/hip_runtime.h>
#include <cstddef>
#include <cstdint>

typedef __attribute__((ext_vector_type(2))) float v2f;
typedef __attribute__((ext_vector_type(8))) float v8f;
typedef __attribute__((ext_vector_type(4))) unsigned int v4u;
typedef __attribute__((ext_vector_type(8))) int v8i;
typedef __attribute__((ext_vector_type(4))) int v4i;

#define NB   4
#define NPTS 256
#define DF   128
#define HF   256

// D = A(16x4) * B(4x16) + C, fp32 WMMA (wave32).
static __device__ __forceinline__ v8f wmma_f32_16x16x4(v2f a, v2f b, v8f c) {
  return __builtin_amdgcn_wmma_f32_16x16x4_f32(false, a, false, b, (short)0, c,
                                               false, false);
}

// ---------------------------------------------------------------------------
// Tensor Data Mover: DMA a contiguous (nrows x ncols) fp32 tile from global
// memory into LDS. D# per CDNA5 ISA ch.8: group0 = {count/type, lds_addr,
// global_addr}, group1 = {data_size, tensor dims, tile dims, stride}.
// Issued by ONE wave; completion via TENSORcnt. 6-arg builtin (clang-23).
// ---------------------------------------------------------------------------
static __device__ __forceinline__ void tdm_load_tile(const float* gsrc,
                                                     unsigned lds_byte_off,
                                                     int nrows, int ncols) {
  unsigned long long ga = (unsigned long long)(uintptr_t)gsrc;
  v4u g0;
  g0[0] = 1u;                                        // count=1 (valid), user mode
  g0[1] = lds_byte_off;                              // D#.lds_addr
  g0[2] = (unsigned)(ga & 0xffffffffu);              // global_addr[31:0]
  g0[3] = (unsigned)((ga >> 32) & 0x01ffffffu)       // global_addr[56:32]
        | (2u << 30);                                 // type = 2 ("image")
  v8i g1;
  g1[0] = (2 << 16);                                 // data_size = 2 -> 4 bytes
  g1[1] = (ncols & 0xffff) << 16;                    // tensor_dim0[15:0]
  g1[2] = ((ncols >> 16) & 0xffff)                   // tensor_dim0[31:16]
        | ((nrows & 0xffff) << 16);                  // tensor_dim1[15:0]
  g1[3] = ((nrows >> 16) & 0xffff)                   // tensor_dim1[31:16]
        | ((ncols & 0xffff) << 16);                  // tile_dim0
  g1[4] = (nrows & 0xffff);                          // tile_dim1 (tile_dim2 = 0)
  g1[5] = ncols;                                     // tensor_dim0_stride[31:0]
  g1[6] = 0;                                         // stride0 hi / stride1 lo
  g1[7] = 0;
  v4i z4 = {0, 0, 0, 0};                             // groups 2/3 unused (2-D)
  v8i z8 = {0, 0, 0, 0, 0, 0, 0, 0};
  __builtin_amdgcn_tensor_load_to_lds(g0, g1, z4, z4, z8, 0);
}

static __device__ __forceinline__ unsigned lds_off(const void* p) {
  return (unsigned)(uintptr_t)p;                     // low 32 bits = LDS address
}

// ---------------------------------------------------------------------------
// Kernel A: xpre[b,j,:] = x[b,j,:] + sum_i adj[b,i,j] * (x[b,i,:] + e[b,i,j,:])
// adj is uniform across the block -> wave-uniform skip of ~85% of e reads.
// ---------------------------------------------------------------------------
__global__ void k_agg(const float* __restrict__ emb_node,
                      const float* __restrict__ emb_edge,
                      const float* __restrict__ adj,
                      const float* __restrict__ mask,
                      float* __restrict__ xpre) {
  const int bj = blockIdx.x;          // b*NPTS + j
  const int b  = bj >> 8;
  const int j  = bj & 255;
  const int d  = threadIdx.x;         // 0..127
  const float mj = mask[bj];
  float acc = emb_node[bj * DF + d] * mj;
  const float* __restrict__ ebase =
      emb_edge + (size_t)b * NPTS * NPTS * DF + (size_t)j * DF;
  for (int i = 0; i < NPTS; ++i) {
    float a = adj[(b * NPTS + i) * NPTS + j];
    if (a != 0.0f) {
      float mi  = mask[b * NPTS + i];
      float xi  = emb_node[(b * NPTS + i) * DF + d] * mi;
      float eij = ebase[(size_t)i * NPTS * DF + d] * (mi * mj);
      acc += a * (xi + eij);
    }
  }
  xpre[bj * DF + d] = acc;
}

// ---------------------------------------------------------------------------
// Kernel B: node MLP + LayerNorm. 16 rows/block, 4 waves.
// Weight K-chunks staged into LDS by the Tensor Data Mover.
// ---------------------------------------------------------------------------
__global__ void k_node(const float* __restrict__ xpre,
                       const float* __restrict__ emb_node,
                       const float* __restrict__ mask,
                       const float* __restrict__ nW1, const float* __restrict__ nb1,
                       const float* __restrict__ nW2, const float* __restrict__ nb2,
                       const float* __restrict__ gX,  const float* __restrict__ bX,
                       float* __restrict__ xnew, float* __restrict__ out_x) {
  const int tid  = threadIdx.x;
  const int wave = tid >> 5;
  const int lane = tid & 31;
  const int half = lane >> 4;
  const int l16  = lane & 15;
  const int row0 = blockIdx.x * 16;

  __shared__ __align__(16) float sA[16 * 132];   // xpre tile (padded)
  __shared__ __align__(16) float sH[16 * 260];   // hidden (padded)
  __shared__ __align__(16) float sW[16 * 256];   // weight K-chunk (TDM dest)
  __shared__ float sMu[16];
  __shared__ float sRs[16];
  float* sOut = sW;                              // alias after GEMM2, stride 132
  const unsigned sW_off = lds_off(sW);

  for (int idx = tid; idx < 16 * DF; idx += 128) {
    int r = idx >> 7, c = idx & 127;
    sA[r * 132 + c] = xpre[(row0 + r) * DF + c];
  }

  // ---- GEMM1: hidden = xpre @ nW1 ; wave owns cols [wave*64, wave*64+64)
  v8f acc1[4] = {};
  for (int kc = 0; kc < DF; kc += 16) {
    __syncthreads();
    if (wave == 0) {
      tdm_load_tile(&nW1[kc * HF], sW_off, 16, HF);
      __builtin_amdgcn_s_wait_tensorcnt(0);
    }
    __syncthreads();
    for (int kk = 0; kk < 16; kk += 4) {
      v2f a;
      a.x = sA[l16 * 132 + kc + kk + 2 * half];
      a.y = sA[l16 * 132 + kc + kk + 2 * half + 1];
#pragma unroll
      for (int ct = 0; ct < 4; ++ct) {
        int n0 = wave * 64 + ct * 16;
        v2f bv;
        bv.x = sW[(kk + 2 * half) * HF + n0 + l16];
        bv.y = sW[(kk + 2 * half + 1) * HF + n0 + l16];
        acc1[ct] = wmma_f32_16x16x4(a, bv, acc1[ct]);
      }
    }
  }
#pragma unroll
  for (int ct = 0; ct < 4; ++ct) {
    int col = wave * 64 + ct * 16 + l16;
    float bias = nb1[col];
#pragma unroll
    for (int v = 0; v < 8; ++v) {
      int r = v + 8 * half;
      sH[r * 260 + col] = fmaxf(acc1[ct][v] + bias, 0.0f);
    }
  }

  // ---- GEMM2: xnew = hidden @ nW2 ; wave owns cols [wave*32, wave*32+32)
  v8f acc2[2] = {};
  for (int kc = 0; kc < HF; kc += 16) {
    __syncthreads();
    if (wave == 0) {
      tdm_load_tile(&nW2[kc * DF], sW_off, 16, DF);
      __builtin_amdgcn_s_wait_tensorcnt(0);
    }
    __syncthreads();
    for (int kk = 0; kk < 16; kk += 4) {
      v2f a;
      a.x = sH[l16 * 260 + kc + kk + 2 * half];
      a.y = sH[l16 * 260 + kc + kk + 2 * half + 1];
#pragma unroll
      for (int ct = 0; ct < 2; ++ct) {
        int n0 = wave * 32 + ct * 16;
        v2f bv;
        bv.x = sW[(kk + 2 * half) * DF + n0 + l16];
        bv.y = sW[(kk + 2 * half + 1) * DF + n0 + l16];
        acc2[ct] = wmma_f32_16x16x4(a, bv, acc2[ct]);
      }
    }
  }
  __syncthreads();
  // epilogue: store xnew, build residual (x + xnew) in LDS
#pragma unroll
  for (int ct = 0; ct < 2; ++ct) {
    int col = wave * 32 + ct * 16 + l16;
    float bias = nb2[col];
#pragma unroll
    for (int v = 0; v < 8; ++v) {
      int r  = v + 8 * half;
      int gr = row0 + r;
      float xn = acc2[ct][v] + bias;
      xnew[gr * DF + col] = xn;
      float xm = emb_node[gr * DF + col] * mask[gr];
      sOut[r * 132 + col] = xm + xn;
    }
  }
  __syncthreads();
  if (tid < 16) {
    float s = 0.f, ss = 0.f;
    for (int c = 0; c < DF; ++c) {
      float v = sOut[tid * 132 + c];
      s += v; ss += v * v;
    }
    float mu  = s * (1.0f / DF);
    float var = ss * (1.0f / DF) - mu * mu;
    sMu[tid] = mu;
    sRs[tid] = rsqrtf(var + 1e-5f);
  }
  __syncthreads();
  for (int idx = tid; idx < 16 * DF; idx += 128) {
    int r = idx >> 7, c = idx & 127;
    out_x[(row0 + r) * DF + c] =
        (sOut[r * 132 + c] - sMu[r]) * sRs[r] * gX[c] + bX[c];
  }
}

// ---------------------------------------------------------------------------
// Kernel C: edge MLP + residual + LayerNorm, fully fused.
// Rows (b,i,j0..j0+15): comb = [xnew_i (shared), xnew_j, e_ij].
// ---------------------------------------------------------------------------
__global__ void k_edge(const float* __restrict__ xnew,
                       const float* __restrict__ emb_edge,
                       const float* __restrict__ mask,
                       const float* __restrict__ eW1, const float* __restrict__ eb1,
                       const float* __restrict__ eW2, const float* __restrict__ eb2,
                       const float* __restrict__ gE,  const float* __restrict__ bE,
                       float* __restrict__ out_e) {
  const int tid  = threadIdx.x;
  const int wave = tid >> 5;
  const int lane = tid & 31;
  const int half = lane >> 4;
  const int l16  = lane & 15;
  const int tile = blockIdx.x;                 // 16384 tiles
  const int j0   = (tile & 15) * 16;
  const int rest = tile >> 4;
  const int i    = rest & 255;
  const int b    = rest >> 8;
  const size_t row_base = (size_t)tile * 16;   // flat (b*N+i)*N + j0

  __shared__ __align__(16) float sC[16 * 388]; // comb tile 16 x 384 (padded)
  __shared__ __align__(16) float sH[16 * 260]; // hidden 16 x 256 (padded)
  __shared__ __align__(16) float sW[16 * 256]; // weight K-chunk (TDM dest)
  __shared__ float sMu[16];
  __shared__ float sRs[16];
  float* sOut = sW;                            // alias, stride 132
  const unsigned sW_off = lds_off(sW);

  const float mi = mask[b * NPTS + i];
  const float* __restrict__ xrow_i = xnew + (b * NPTS + i) * DF;
  for (int idx = tid; idx < 16 * DF; idx += 128) {
    int r = idx >> 7, c = idx & 127;
    int jn = b * NPTS + j0 + r;
    sC[r * 388 + c]        = xrow_i[c];
    sC[r * 388 + 128 + c]  = xnew[jn * DF + c];
    sC[r * 388 + 256 + c]  = emb_edge[(row_base + r) * DF + c] * (mi * mask[jn]);
  }

  // ---- GEMM1: hidden = comb @ eW1 (K=384); wave owns cols [wave*64, +64)
  v8f acc1[4] = {};
  for (int kc = 0; kc < 3 * DF; kc += 16) {
    __syncthreads();
    if (wave == 0) {
      tdm_load_tile(&eW1[kc * HF], sW_off, 16, HF);
      __builtin_amdgcn_s_wait_tensorcnt(0);
    }
    __syncthreads();
    for (int kk = 0; kk < 16; kk += 4) {
      v2f a;
      a.x = sC[l16 * 388 + kc + kk + 2 * half];
      a.y = sC[l16 * 388 + kc + kk + 2 * half + 1];
#pragma unroll
      for (int ct = 0; ct < 4; ++ct) {
        int n0 = wave * 64 + ct * 16;
        v2f bv;
        bv.x = sW[(kk + 2 * half) * HF + n0 + l16];
        bv.y = sW[(kk + 2 * half + 1) * HF + n0 + l16];
        acc1[ct] = wmma_f32_16x16x4(a, bv, acc1[ct]);
      }
    }
  }
#pragma unroll
  for (int ct = 0; ct < 4; ++ct) {
    int col = wave * 64 + ct * 16 + l16;
    float bias = eb1[col];
#pragma unroll
    for (int v = 0; v < 8; ++v) {
      int r = v + 8 * half;
      sH[r * 260 + col] = fmaxf(acc1[ct][v] + bias, 0.0f);
    }
  }

  // ---- GEMM2: e_new = hidden @ eW2 (K=256); wave owns cols [wave*32, +32)
  v8f acc2[2] = {};
  for (int kc = 0; kc < HF; kc += 16) {
    __syncthreads();
    if (wave == 0) {
      tdm_load_tile(&eW2[kc * DF], sW_off, 16, DF);
      __builtin_amdgcn_s_wait_tensorcnt(0);
    }
    __syncthreads();
    for (int kk = 0; kk < 16; kk += 4) {
      v2f a;
      a.x = sH[l16 * 260 + kc + kk + 2 * half];
      a.y = sH[l16 * 260 + kc + kk + 2 * half + 1];
#pragma unroll
      for (int ct = 0; ct < 2; ++ct) {
        int n0 = wave * 32 + ct * 16;
        v2f bv;
        bv.x = sW[(kk + 2 * half) * DF + n0 + l16];
        bv.y = sW[(kk + 2 * half + 1) * DF + n0 + l16];
        acc2[ct] = wmma_f32_16x16x4(a, bv, acc2[ct]);
      }
    }
  }
  __syncthreads();
  // epilogue: e + e_new into LDS (e residual still resident in sC)
#pragma unroll
  for (int ct = 0; ct < 2; ++ct) {
    int col = wave * 32 + ct * 16 + l16;
    float bias = eb2[col];
#pragma unroll
    for (int v = 0; v < 8; ++v) {
      int r = v + 8 * half;
      float en = acc2[ct][v] + bias;
      sOut[r * 132 + col] = sC[r * 388 + 256 + col] + en;
    }
  }
  __syncthreads();
  if (tid < 16) {
    float s = 0.f, ss = 0.f;
    for (int c = 0; c < DF; ++c) {
      float v = sOut[tid * 132 + c];
      s += v; ss += v * v;
    }
    float mu  = s * (1.0f / DF);
    float var = ss * (1.0f / DF) - mu * mu;
    sMu[tid] = mu;
    sRs[tid] = rsqrtf(var + 1e-5f);
  }
  __syncthreads();
  for (int idx = tid; idx < 16 * DF; idx += 128) {
    int r = idx >> 7, c = idx & 127;
    out_e[(row_base + r) * DF + c] =
        (sOut[r * 132 + c] - sMu[r]) * sRs[r] * gE[c] + bE[c];
  }
}

// ---------------------------------------------------------------------------
extern "C" void kernel_launch(void* const* d_in, const int* in_sizes, int n_in,
                              void* d_out, int out_size, void* d_ws, size_t ws_size,
                              hipStream_t stream) {
  (void)in_sizes; (void)n_in; (void)out_size; (void)ws_size;
  const float* emb_node  = (const float*)d_in[0];
  const float* emb_edge  = (const float*)d_in[1];
  const float* adj       = (const float*)d_in[2];
  const float* node_mask = (const float*)d_in[3];
  const float* nW1 = (const float*)d_in[4];
  const float* nb1 = (const float*)d_in[5];
  const float* nW2 = (const float*)d_in[6];
  const float* nb2 = (const float*)d_in[7];
  const float* eW1 = (const float*)d_in[8];
  const float* eb1 = (const float*)d_in[9];
  const float* eW2 = (const float*)d_in[10];
  const float* eb2 = (const float*)d_in[11];
  const float* gX  = (const float*)d_in[12];
  const float* bX  = (const float*)d_in[13];
  const float* gE  = (const float*)d_in[14];
  const float* bE  = (const float*)d_in[15];

  float* ws_xpre = (float*)d_ws;                 // B*N*D floats
  float* ws_xnew = ws_xpre + NB * NPTS * DF;     // B*N*D floats
  float* out_x   = (float*)d_out;
  float* out_e   = out_x + NB * NPTS * DF;

  k_agg<<<NB * NPTS, DF, 0, stream>>>(emb_node, emb_edge, adj, node_mask, ws_xpre);
  k_node<<<(NB * NPTS) / 16, 128, 0, stream>>>(ws_xpre, emb_node, node_mask,
                                               nW1, nb1, nW2, nb2, gX, bX,
                                               ws_xnew, out_x);
  k_edge<<<(NB * NPTS * NPTS) / 16, 128, 0, stream>>>(ws_xnew, emb_edge, node_mask,
                                                      eW1, eb1, eW2, eb2, gE, bE,
                                                      out_e);
}